// VolumeRenderer_61100204753700
// MI455X (gfx1250) — compile-verified
//
#include <hip/hip_runtime.h>
#include <hip/hip_bf16.h>
#include <cstdint>

// ---------------------------------------------------------------------------
// Volume renderer (NeRF-style weighted integration), MI455X / gfx1250.
// Bandwidth-bound: 160 MB in / 1 MB out -> ~6.9 us floor @ 23.3 TB/s.
//   * wave owns 8 rays; ALL 40 async global->LDS B128 copies issued up-front
//     into an 8-slot LDS ring. LDS slots advance in lockstep with the global
//     stride, so every copy is base-VGPR + compile-time IOFFSET (zero per-ray
//     address VALU). Ray i waits s_wait_asynccnt 5*(7-i) (in-order returns).
//   * cumulative optical depth: DPP row_shr scan + ds_swizzle row-broadcast.
//   * transmittance via multiplicative chain: 5 exps/ray instead of 8
//     (residual factor exp2(EPS2) = 1+1e-10, ~600x below f32 ULP).
//   * 4-channel output reduction: channel-folding ds_swizzle xor butterfly.
// ---------------------------------------------------------------------------

#define S_SAMPLES 128
#define C_FEAT 3
#define L2E 1.44269504088896340736f
#define EPS2 1.4426950408889634e-10f  /* EPS * log2(e) */
#define FAR_DELTA 1e10f

#define WAVES_PER_BLOCK 4
#define RAYS_PER_WAVE 8
#define RAYS_PER_BLOCK (WAVES_PER_BLOCK * RAYS_PER_WAVE)
// per-wave ring (floats): feature[8*384]=3072 | density[8*128]=1024 | depth[8*128]=1024
#define WAVE_FLOATS 5120  /* 20 KB */
#define DENS_OFS 3072
#define DEP_OFS 4096

typedef int v4i __attribute__((ext_vector_type(4)));
typedef __attribute__((address_space(1))) v4i* gptr_v4i;
typedef __attribute__((address_space(3))) v4i* lptr_v4i;

// ---- gfx1250 async global->LDS copy; IOFFSET applies to BOTH addresses ----
template <int BYTEOFS>
__device__ __forceinline__ void async_cp128o(const float* g, float* l) {
#if defined(__HIP_DEVICE_COMPILE__)
  __builtin_amdgcn_global_load_async_to_lds_b128(
      (gptr_v4i)(g), (lptr_v4i)(l), BYTEOFS, /*cpol*/ 0);
#endif
}

#if defined(__HIP_DEVICE_COMPILE__)
#define WAIT_ASYNCCNT(n) __builtin_amdgcn_s_wait_asynccnt(n)
#else
#define WAIT_ASYNCCNT(n) ((void)0)
#endif

// ---- single-instruction cross-lane helpers --------------------------------
template <int CTRL, int ROWMASK>
__device__ __forceinline__ float dpp0(float x) {
  return __builtin_bit_cast(
      float, __builtin_amdgcn_update_dpp(0, __builtin_bit_cast(int, x), CTRL,
                                         ROWMASK, 0xf, true));
}
template <int PAT>
__device__ __forceinline__ float swz(float x) {
  return __builtin_bit_cast(
      float, __builtin_amdgcn_ds_swizzle(__builtin_bit_cast(int, x), PAT));
}
#define SWZ_XOR(m) ((m) << 10 | 0x1f)  // group-of-32, xor by m
#define SWZ_BCAST15 (15 << 5)          // every lane reads lane 15

__device__ __forceinline__ float fexp2(float x) {
#if __has_builtin(__builtin_amdgcn_exp2f)
  return __builtin_amdgcn_exp2f(x);
#else
  return exp2f(x);
#endif
}

// ---- per-ray body (I is a compile-time slot index) ------------------------
template <int I>
__device__ __forceinline__ void do_ray(const float* wbuf, float* __restrict__ out,
                                       int ray0, int lane, bool hi16, bool hi8) {
  WAIT_ASYNCCNT(5 * (RAYS_PER_WAVE - 1 - I));  // slot I's 5 copies landed

  const float* fb = wbuf + I * 384 + lane * 12;           // 48B-aligned
  const float* db = wbuf + DENS_OFS + I * 128 + lane * 4;
  const float* pb = wbuf + DEP_OFS + I * 128 + lane * 4;

  float4 den = *reinterpret_cast<const float4*>(db);
  float4 dep = *reinterpret_cast<const float4*>(pb);
  float dnext = pb[(lane < 31) ? 4 : 0];  // depth[4l+4]; unused for lane 31

  // scale density by log2(e) once: all optical depths live in exp2 domain
  den.x *= L2E; den.y *= L2E; den.z *= L2E; den.w *= L2E;

  float d0 = dep.y - dep.x;
  float d1 = dep.z - dep.y;
  float d2 = dep.w - dep.z;
  float d3 = (lane == 31) ? FAR_DELTA : (dnext - dep.w);

  float e0 = den.x * d0, e1 = den.y * d1, e2 = den.z * d2, e3 = den.w * d3;

  // wave32 exclusive scan: DPP row_shr within 16-lane rows, then ds_swizzle
  // broadcast of lane 15 (row0 total) added into the upper row.
  float lsum = e0 + e1 + e2 + e3;
  float inc = lsum;
  inc += dpp0<0x111, 0xf>(inc);  // row_shr:1
  inc += dpp0<0x112, 0xf>(inc);  // row_shr:2
  inc += dpp0<0x114, 0xf>(inc);  // row_shr:4
  inc += dpp0<0x118, 0xf>(inc);  // row_shr:8
  float row0tot = swz<SWZ_BCAST15>(inc);
  inc += hi16 ? row0tot : 0.0f;
  float base = inc - lsum;  // log2-domain optical depth before sample 4l

  // Per-sample survival A_j = exp2(EPS2 - e_j) (independent -> TRANS pipe),
  // transmittance chain T_{j+1} = T_j * A_j, weight w_j = T_j - T_{j+1}.
  float A0 = fexp2(EPS2 - e0);
  float A1 = fexp2(EPS2 - e1);
  float A2 = fexp2(EPS2 - e2);
  float A3 = fexp2(EPS2 - e3);
  float T0 = fexp2(EPS2 - base);
  float T1 = T0 * A0; float w0 = T0 - T1;
  float T2 = T1 * A1; float w1 = T1 - T2;
  float T3 = T2 * A2; float w2 = T2 - T3;
  float T4 = T3 * A3; float w3 = T3 - T4;

  float4 fA = *reinterpret_cast<const float4*>(fb);
  float4 fB = *reinterpret_cast<const float4*>(fb + 4);
  float4 fC = *reinterpret_cast<const float4*>(fb + 8);

  float a0 = w0 * fA.x + w1 * fA.w + w2 * fB.z + w3 * fC.y;      // ch 0
  float a1 = w0 * fA.y + w1 * fB.x + w2 * fB.w + w3 * fC.z;      // ch 1
  float a2 = w0 * fA.z + w1 * fB.y + w2 * fC.x + w3 * fC.w;      // ch 2
  float a3 = w0 * dep.x + w1 * dep.y + w2 * dep.z + w3 * dep.w;  // depth

  // channel-folding reduction (6 ds_swizzle total); channel = lane>>3
  float v0 = (hi16 ? a2 : a0) + swz<SWZ_XOR(16)>(hi16 ? a0 : a2);
  float v1 = (hi16 ? a3 : a1) + swz<SWZ_XOR(16)>(hi16 ? a1 : a3);
  float w8 = (hi8 ? v1 : v0) + swz<SWZ_XOR(8)>(hi8 ? v0 : v1);
  w8 += swz<SWZ_XOR(4)>(w8);
  w8 += swz<SWZ_XOR(2)>(w8);
  w8 += swz<SWZ_XOR(1)>(w8);

  if ((lane & 7) == 0) {
    out[(size_t)(ray0 + I) * 4 + (lane >> 3)] = w8;
  }
}

__global__ __launch_bounds__(WAVES_PER_BLOCK * 32) void
VolumeRenderer_61100204753700_kernel(const float* __restrict__ density,
                                     const float* __restrict__ feature,
                                     const float* __restrict__ depth,
                                     float* __restrict__ out, int N) {
  __shared__ float smem[WAVES_PER_BLOCK * WAVE_FLOATS];  // 80 KB

  const int lane = threadIdx.x & 31;
  const int wave = threadIdx.x >> 5;
  float* wbuf = smem + wave * WAVE_FLOATS;

  const bool hi16 = (lane & 16) != 0;
  const bool hi8  = (lane & 8) != 0;

  int ray0 = (blockIdx.x * WAVES_PER_BLOCK + wave) * RAYS_PER_WAVE;
  if (ray0 + RAYS_PER_WAVE > N) {
    ray0 = N - RAYS_PER_WAVE;   // tail waves recompute a few rays (idempotent)
    if (ray0 < 0) return;       // N < 8
  }

  // Fixed per-lane base addresses; every copy below differs only by IOFFSET.
  float* lf = wbuf + lane * 4;
  float* ld = wbuf + DENS_OFS + lane * 4;
  float* lp = wbuf + DEP_OFS + lane * 4;
  const float* gf = feature + (size_t)ray0 * (S_SAMPLES * C_FEAT) + lane * 4;
  const float* gd = density + (size_t)ray0 * S_SAMPLES + lane * 4;
  const float* gp = depth   + (size_t)ray0 * S_SAMPLES + lane * 4;

  // Issue ALL 40 async B128 copies (20 KB/wave in flight, in-order returns).
#define ISSUE_RAY(I)                                   \
  async_cp128o<(I) * 1536>(gf, lf);                    \
  async_cp128o<(I) * 1536 + 512>(gf, lf);              \
  async_cp128o<(I) * 1536 + 1024>(gf, lf);             \
  async_cp128o<(I) * 512>(gd, ld);                     \
  async_cp128o<(I) * 512>(gp, lp);
  ISSUE_RAY(0) ISSUE_RAY(1) ISSUE_RAY(2) ISSUE_RAY(3)
  ISSUE_RAY(4) ISSUE_RAY(5) ISSUE_RAY(6) ISSUE_RAY(7)
#undef ISSUE_RAY

  do_ray<0>(wbuf, out, ray0, lane, hi16, hi8);
  do_ray<1>(wbuf, out, ray0, lane, hi16, hi8);
  do_ray<2>(wbuf, out, ray0, lane, hi16, hi8);
  do_ray<3>(wbuf, out, ray0, lane, hi16, hi8);
  do_ray<4>(wbuf, out, ray0, lane, hi16, hi8);
  do_ray<5>(wbuf, out, ray0, lane, hi16, hi8);
  do_ray<6>(wbuf, out, ray0, lane, hi16, hi8);
  do_ray<7>(wbuf, out, ray0, lane, hi16, hi8);
}

extern "C" void kernel_launch(void* const* d_in, const int* in_sizes, int n_in,
                              void* d_out, int out_size, void* d_ws, size_t ws_size,
                              hipStream_t stream) {
  const float* density = (const float*)d_in[0];  // [N, 128]
  const float* feature = (const float*)d_in[1];  // [N, 128, 3]
  const float* depth   = (const float*)d_in[2];  // [N, 128]
  float* out = (float*)d_out;                    // [N, 4]

  const int N = in_sizes[0] / S_SAMPLES;
  const int blocks = (N + RAYS_PER_BLOCK - 1) / RAYS_PER_BLOCK;
  VolumeRenderer_61100204753700_kernel<<<blocks, WAVES_PER_BLOCK * 32, 0, stream>>>(
      density, feature, depth, out, N);
}